// KN_A_23201413333623
// MI455X (gfx1250) — compile-verified
//
#include <hip/hip_runtime.h>
#include <hip/hip_bf16.h>
#include <math.h>

typedef float v8f __attribute__((ext_vector_type(8)));
typedef float v2f __attribute__((ext_vector_type(2)));

#define NVOX 32768   // 32^3

// ---------------------------------------------------------------------------
// Kernel A: x_feat[s] = sum_c ((x1-x2)^2 * mask_ds)[c,s] * W[c] + b
// One wave computes 16 voxels via 16 chained V_WMMA_F32_16X16X4_F32 (K=64).
// A layout (16x4 f32, documented): lanes 0-15 -> M=lane, VGPR0=K0/VGPR1=K1;
// lanes 16-31 -> M=lane-16, VGPR0=K2/VGPR1=K3.
// B is replicated across all N columns (B[k][n] = W[c0+k] for all n), so all
// D columns are equal; extract D[:,*] from lanes 0 (M=0..7) and 16 (M=8..15).
// ---------------------------------------------------------------------------
__global__ void __launch_bounds__(256) kA_feat_wmma(
    const float* __restrict__ x1, const float* __restrict__ x2,
    const float* __restrict__ mask, const float* __restrict__ W,
    const float* __restrict__ bias, float* __restrict__ xfeat)
{
  const int lane  = threadIdx.x & 31;
  const int wid   = blockIdx.x * (blockDim.x >> 5) + (threadIdx.x >> 5);
  const int s0    = wid * 16;
  const int m     = lane & 15;   // M row this lane-half owns
  const int khalf = lane >> 4;   // 0: K0/K1, 1: K2/K3
  const int s     = s0 + m;
  const int z = s >> 10, y = (s >> 5) & 31, x = s & 31;
  const float mval = mask[(z * 8) * 65536 + (y * 8) * 256 + (x * 8)];

  v8f acc = {0.f, 0.f, 0.f, 0.f, 0.f, 0.f, 0.f, 0.f};
#pragma unroll
  for (int k = 0; k < 16; ++k) {
    const int c0 = k * 4 + khalf * 2;
    float d0 = x1[c0 * NVOX + s]       - x2[c0 * NVOX + s];
    float d1 = x1[(c0 + 1) * NVOX + s] - x2[(c0 + 1) * NVOX + s];
    v2f a;    a.x = d0 * d0 * mval; a.y = d1 * d1 * mval;
    v2f bm;   bm.x = W[c0];         bm.y = W[c0 + 1];   // same for every N
    acc = __builtin_amdgcn_wmma_f32_16x16x4_f32(
        /*neg_a=*/false, a, /*neg_b=*/false, bm,
        /*c_mod=*/(short)0, acc, /*reuse_a=*/false, /*reuse_b=*/false);
  }
  const float bv = bias[0];
  if (m == 0) {
#pragma unroll
    for (int r = 0; r < 8; ++r)
      xfeat[s0 + khalf * 8 + r] = acc[r] + bv;
  }
}

// ---------------------------------------------------------------------------
// Kernel B: candidate flags = eroded(mask_ds) && (maxpool5(x_feat) == x_feat)
// ---------------------------------------------------------------------------
__global__ void __launch_bounds__(256) kB_cand(
    const float* __restrict__ xfeat, const float* __restrict__ mask,
    int* __restrict__ cand)
{
  int s = blockIdx.x * blockDim.x + threadIdx.x;
  if (s >= NVOX) return;
  int z = s >> 10, y = (s >> 5) & 31, x = s & 31;
  float v = xfeat[s];
  float maxv = -INFINITY;
  for (int dz = -2; dz <= 2; ++dz) {
    int zz = z + dz; if ((unsigned)zz >= 32u) continue;
    for (int dy = -2; dy <= 2; ++dy) {
      int yy = y + dy; if ((unsigned)yy >= 32u) continue;
      for (int dx = -2; dx <= 2; ++dx) {
        int xx = x + dx; if ((unsigned)xx >= 32u) continue;
        float t = xfeat[(zz << 10) + (yy << 5) + xx];
        maxv = fmaxf(maxv, t);
      }
    }
  }
  // erosion: conv of (1-mask) with 6-neighborhood SE, zero padding
  float cnt = 0.f;
  const int off[6][3] = {{-1,0,0},{1,0,0},{0,-1,0},{0,1,0},{0,0,-1},{0,0,1}};
  for (int i = 0; i < 6; ++i) {
    int zz = z + off[i][0], yy = y + off[i][1], xx = x + off[i][2];
    if ((unsigned)zz < 32u && (unsigned)yy < 32u && (unsigned)xx < 32u)
      cnt += 1.0f - mask[(zz * 8) * 65536 + (yy * 8) * 256 + (xx * 8)];
  }
  int eroded = (1.0f - fminf(fmaxf(cnt, 0.f), 1.f)) > 0.5f;
  cand[s] = (eroded && (maxv == v)) ? 1 : 0;
}

// ---------------------------------------------------------------------------
// Kernel C: farthest-point sampling, 16 points, jnp.argmax first-index
// tie-breaking. Single workgroup of 1024 threads; dist lives in global ws.
// kpts out: [0..15]=z_norm, [16..31]=y_norm, [32..47]=x_norm
// ---------------------------------------------------------------------------
__global__ void __launch_bounds__(1024) kC_fps(
    const int* __restrict__ cand, float* __restrict__ dist,
    float* __restrict__ kpts)
{
  __shared__ float sval[1024];
  __shared__ int   sidx[1024];
  const int tid = threadIdx.x;

  // i0 = first candidate index (argmax of boolean)
  int best = 1 << 30;
  for (int s = tid; s < NVOX; s += 1024)
    if (cand[s]) { best = s; break; }
  sidx[tid] = best;
  __syncthreads();
  for (int off = 512; off > 0; off >>= 1) {
    if (tid < off) sidx[tid] = min(sidx[tid], sidx[tid + off]);
    __syncthreads();
  }
  int sel = sidx[0];
  if (sel >= NVOX) sel = 0;   // no candidates -> argmax returns 0
  __syncthreads();

  int selz = sel >> 10, sely = (sel >> 5) & 31, selx = sel & 31;
  for (int s = tid; s < NVOX; s += 1024) {
    float dz = (float)((s >> 10) - selz);
    float dy = (float)(((s >> 5) & 31) - sely);
    float dx = (float)((s & 31) - selx);
    dist[s] = dz * dz + dy * dy + dx * dx;
  }
  if (tid == 0) {
    kpts[0]  = (float)selz * (2.0f / 32.0f) - 1.0f;
    kpts[16] = (float)sely * (2.0f / 32.0f) - 1.0f;
    kpts[32] = (float)selx * (2.0f / 32.0f) - 1.0f;
  }
  __syncthreads();

  for (int it = 1; it < 16; ++it) {
    float bv = -2.0f; int bi = 0;
    for (int s = tid; s < NVOX; s += 1024) {
      float val = cand[s] ? dist[s] : -1.0f;
      if (val > bv) { bv = val; bi = s; }   // strictly > keeps earliest index
    }
    sval[tid] = bv; sidx[tid] = bi;
    __syncthreads();
    for (int off = 512; off > 0; off >>= 1) {
      if (tid < off) {
        float v2 = sval[tid + off]; int i2 = sidx[tid + off];
        if (v2 > sval[tid] || (v2 == sval[tid] && i2 < sidx[tid])) {
          sval[tid] = v2; sidx[tid] = i2;
        }
      }
      __syncthreads();
    }
    sel  = sidx[0];
    selz = sel >> 10; sely = (sel >> 5) & 31; selx = sel & 31;
    if (tid == 0) {
      kpts[it]      = (float)selz * (2.0f / 32.0f) - 1.0f;
      kpts[16 + it] = (float)sely * (2.0f / 32.0f) - 1.0f;
      kpts[32 + it] = (float)selx * (2.0f / 32.0f) - 1.0f;
    }
    __syncthreads();
    for (int s = tid; s < NVOX; s += 1024) {
      float dz = (float)((s >> 10) - selz);
      float dy = (float)(((s >> 5) & 31) - sely);
      float dx = (float)((s & 31) - selx);
      float nd = dz * dz + dy * dy + dx * dx;
      dist[s] = fminf(dist[s], nd);
    }
    __syncthreads();
  }
}

// ---------------------------------------------------------------------------
// Kernel D: gauss_mu. One block per map m: weighted centroid of
// relu(x_feat)*attn(kpt_m)*mask_ds. gmu[axis*16+m], axis order z,y,x.
// ---------------------------------------------------------------------------
__global__ void __launch_bounds__(256) kD_mu(
    const float* __restrict__ xfeat, const float* __restrict__ mask,
    const float* __restrict__ kpts, float* __restrict__ gmu)
{
  const int m = blockIdx.x, tid = threadIdx.x;
  const float mz = kpts[m], my = kpts[16 + m], mx = kpts[32 + m];
  const float step = 2.0f / 31.0f;
  float S = 0.f, Sz = 0.f, Sy = 0.f, Sx = 0.f;
  for (int s = tid; s < NVOX; s += 256) {
    int z = s >> 10, y = (s >> 5) & 31, x = s & 31;
    float zl = z * step - 1.0f, yl = y * step - 1.0f, xl = x * step - 1.0f;
    float dz = zl - mz, dy = yl - my, dx = xl - mx;
    float attn = __expf(-100.0f * (dz * dz + dy * dy + dx * dx));
    float f = xfeat[s]; f = f > 0.f ? f : 0.f;
    float mval = mask[(z * 8) * 65536 + (y * 8) * 256 + (x * 8)];
    float p = f * attn * mval;
    S += p; Sz += p * zl; Sy += p * yl; Sx += p * xl;
  }
  __shared__ float rs[256], rz[256], ry[256], rx[256];
  rs[tid] = S; rz[tid] = Sz; ry[tid] = Sy; rx[tid] = Sx;
  __syncthreads();
  for (int off = 128; off > 0; off >>= 1) {
    if (tid < off) {
      rs[tid] += rs[tid + off]; rz[tid] += rz[tid + off];
      ry[tid] += ry[tid + off]; rx[tid] += rx[tid + off];
    }
    __syncthreads();
  }
  if (tid == 0) {
    float denom = rs[0] * (1.0f / 1024.0f) + 1e-8f;
    gmu[0 * 16 + m] = (rz[0] * (1.0f / 1024.0f)) / denom;
    gmu[1 * 16 + m] = (ry[0] * (1.0f / 1024.0f)) / denom;
    gmu[2 * 16 + m] = (rx[0] * (1.0f / 1024.0f)) / denom;
  }
}

// ---------------------------------------------------------------------------
// Kernel E: gaussian maps, 16 maps of n^3 (n = 32/64/128). Store-BW bound:
// 4 voxels (same row) per thread, one float4 store, v_exp_f32 per voxel.
// ---------------------------------------------------------------------------
__global__ void __launch_bounds__(256) kE_gmap(
    const float* __restrict__ gmu, float* __restrict__ outp,
    int n, float step)
{
  int idx = blockIdx.x * blockDim.x + threadIdx.x;
  int n2 = n * n, n3 = n2 * n;
  int total4 = 16 * n3 / 4;
  if (idx >= total4) return;
  int base = idx * 4;
  int m  = base / n3;
  int r  = base - m * n3;
  int z  = r / n2;
  int r2 = r - z * n2;
  int y  = r2 / n;
  int x0 = r2 - y * n;           // multiple of 4, same z/y for all 4
  float mz = gmu[m], my = gmu[16 + m], mx = gmu[32 + m];
  float dz = z * step - 1.0f - mz;
  float dy = y * step - 1.0f - my;
  float b2 = dz * dz + dy * dy;
  float4 o;
  float dx;
  dx = (x0 + 0) * step - 1.0f - mx; o.x = __expf(-100.0f * (b2 + dx * dx));
  dx = (x0 + 1) * step - 1.0f - mx; o.y = __expf(-100.0f * (b2 + dx * dx));
  dx = (x0 + 2) * step - 1.0f - mx; o.z = __expf(-100.0f * (b2 + dx * dx));
  dx = (x0 + 3) * step - 1.0f - mx; o.w = __expf(-100.0f * (b2 + dx * dx));
  *(float4*)(outp + base) = o;
}

// ---------------------------------------------------------------------------
extern "C" void kernel_launch(void* const* d_in, const int* in_sizes, int n_in,
                              void* d_out, int out_size, void* d_ws, size_t ws_size,
                              hipStream_t stream)
{
  const float* x1   = (const float*)d_in[0];   // (1,64,32,32,32)
  const float* x2   = (const float*)d_in[1];   // (1,64,32,32,32)
  const float* mask = (const float*)d_in[2];   // (1,1,256,256,256)
  const float* W    = (const float*)d_in[3];   // (1,64)
  const float* bias = (const float*)d_in[4];   // (1,)

  float* out   = (float*)d_out;
  float* xfeat = out;                              // 32768
  float* gmu   = out + 32768;                      // 48
  float* g0    = out + 32768 + 48;                 // 16*32^3
  float* g1    = g0 + 16 * 32768;                  // 16*64^3
  float* g2    = g1 + 16 * 262144;                 // 16*128^3

  int*   cand = (int*)d_ws;
  float* dist = (float*)((char*)d_ws + NVOX * sizeof(int));
  float* kpts = (float*)((char*)d_ws + 2 * NVOX * sizeof(int));

  // A: x_feat via WMMA — 2048 waves of 16 voxels each
  kA_feat_wmma<<<256, 256, 0, stream>>>(x1, x2, mask, W, bias, xfeat);
  // B: candidate flags
  kB_cand<<<(NVOX + 255) / 256, 256, 0, stream>>>(xfeat, mask, cand);
  // C: farthest-point sampling (sequential, single workgroup)
  kC_fps<<<1, 1024, 0, stream>>>(cand, dist, kpts);
  // D: gauss_mu (16 maps)
  kD_mu<<<16, 256, 0, stream>>>(xfeat, mask, kpts, gmu);
  // E: gaussian maps at 32^3, 64^3, 128^3
  {
    int t4;
    t4 = 16 * 32 * 32 * 32 / 4;
    kE_gmap<<<(t4 + 255) / 256, 256, 0, stream>>>(gmu, g0, 32, 2.0f / 31.0f);
    t4 = 16 * 64 * 64 * 64 / 4;
    kE_gmap<<<(t4 + 255) / 256, 256, 0, stream>>>(gmu, g1, 64, 2.0f / 63.0f);
    t4 = 16 * 128 * 128 * 128 / 4;
    kE_gmap<<<(t4 + 255) / 256, 256, 0, stream>>>(gmu, g2, 128, 2.0f / 127.0f);
  }
}